// WESUP_53790170415513
// MI455X (gfx1250) — compile-verified
//
#include <hip/hip_runtime.h>
#include <hip/hip_bf16.h>
#include <math.h>

#define NUM_SEG 533
#define C_FEAT  2112
#define HW      160000
#define M_PAD   544      // 34 tiles of 16
#define M_TILES 34

typedef __attribute__((ext_vector_type(2))) float v2f;
typedef __attribute__((ext_vector_type(8))) float v8f;

// ---------------- workspace layout (float units) ----------------
#define OFF_SUMS 0                               // 533*2112
#define OFF_CNT  (OFF_SUMS + NUM_SEG*C_FEAT)     // 544
#define OFF_LABC (OFF_CNT + 544)                 // 1600 (533*3 padded)
#define OFF_MEAN (OFF_LABC + 1600)               // 544*2112 (zero-padded rows)
#define OFF_NRM  (OFF_MEAN + M_PAD*C_FEAT)       // 544*2112
#define OFF_H1   (OFF_NRM + M_PAD*C_FEAT)        // 544*1024
#define OFF_H2   (OFF_H1 + M_PAD*1024)           // 544*1024
#define OFF_H3   (OFF_H2 + M_PAD*1024)           // 544*32
#define OFF_AFF  (OFF_H3 + M_PAD*32)             // 544*544
#define OFF_BEST (OFF_AFF + M_PAD*M_PAD)         // 544
#define OFF_BIDX (OFF_BEST + 544)                // 544 (int)
#define OFF_LAB  (OFF_BIDX + 544)                // 544 (int)
#define OFF_LBD  (OFF_LAB + 544)                 // 544 (int)
#define WS_FLOATS (OFF_LBD + 544)

// ---------------- zero init of accumulators ----------------
__global__ void wesup_zero(float* ws, int n) {
    int i = blockIdx.x * blockDim.x + threadIdx.x;
    if (i < n) ws[i] = 0.0f;
}

// ---------------- segment sum of features (LDS-binned) ----------------
// grid.x = C_FEAT/8 channel groups, grid.y = pixel chunks
#define CH_PER_BLK 8
#define PIX_CHUNKS 4
__global__ void wesup_segsum(const float* __restrict__ fm, const int* __restrict__ sp,
                             float* __restrict__ sums) {
    __shared__ float bins[CH_PER_BLK * NUM_SEG];
    const int tid = threadIdx.x;
    for (int i = tid; i < CH_PER_BLK * NUM_SEG; i += blockDim.x) bins[i] = 0.0f;
    __syncthreads();

    const int cbase = blockIdx.x * CH_PER_BLK;
    const int chunk = (HW + PIX_CHUNKS - 1) / PIX_CHUNKS;
    const int p0 = blockIdx.y * chunk;
    const int p1 = min(p0 + chunk, HW);

    for (int p = p0 + tid; p < p1; p += blockDim.x) {
        const int s = sp[p];
        const int pnext = p + 256;
        if (pnext < p1) {
            // prime the 23.3 TB/s stream one trip ahead (global_prefetch_b8)
            #pragma unroll
            for (int c = 0; c < CH_PER_BLK; ++c)
                __builtin_prefetch(&fm[(size_t)(cbase + c) * HW + pnext], 0, 0);
        }
        #pragma unroll
        for (int c = 0; c < CH_PER_BLK; ++c) {
            float v = fm[(size_t)(cbase + c) * HW + p];
            atomicAdd(&bins[c * NUM_SEG + s], v);
        }
    }
    __syncthreads();
    for (int i = tid; i < CH_PER_BLK * NUM_SEG; i += blockDim.x) {
        float v = bins[i];
        if (v != 0.0f) {
            int c = i / NUM_SEG, s = i % NUM_SEG;
            atomicAdd(&sums[(size_t)s * C_FEAT + cbase + c], v);
        }
    }
}

// ---------------- pixel counts + label vote counts ----------------
__global__ void wesup_counts(const int* __restrict__ sp, const int* __restrict__ y,
                             float* __restrict__ cnt, float* __restrict__ labc) {
    __shared__ float lcnt[NUM_SEG];
    __shared__ float llab[NUM_SEG * 3];
    const int tid = threadIdx.x;
    for (int i = tid; i < NUM_SEG; i += blockDim.x) lcnt[i] = 0.0f;
    for (int i = tid; i < NUM_SEG * 3; i += blockDim.x) llab[i] = 0.0f;
    __syncthreads();
    for (int p = blockIdx.x * blockDim.x + tid; p < HW; p += gridDim.x * blockDim.x) {
        int s = sp[p];
        atomicAdd(&lcnt[s], 1.0f);
        atomicAdd(&llab[s * 3 + y[p]], 1.0f);
    }
    __syncthreads();
    for (int i = tid; i < NUM_SEG; i += blockDim.x)
        if (lcnt[i] != 0.0f) atomicAdd(&cnt[i], lcnt[i]);
    for (int i = tid; i < NUM_SEG * 3; i += blockDim.x)
        if (llab[i] != 0.0f) atomicAdd(&labc[i], llab[i]);
}

// ---------------- mean features (padded to 544 rows, zero pad) ----------------
__global__ void wesup_mean(const float* __restrict__ sums, const float* __restrict__ cnt,
                           float* __restrict__ mean) {
    int i = blockIdx.x * blockDim.x + threadIdx.x;
    if (i >= M_PAD * C_FEAT) return;
    int row = i / C_FEAT;
    float v = 0.0f;
    if (row < NUM_SEG) v = sums[i] / fmaxf(cnt[row], 1.0f);
    mean[i] = v;
}

// ---------------- majority label + tie-break ----------------
__global__ void wesup_labels(const float* __restrict__ labc, int* __restrict__ lab,
                             int* __restrict__ labeled) {
    int s = blockIdx.x * blockDim.x + threadIdx.x;
    if (s >= NUM_SEG) return;
    float c0 = labc[s * 3 + 0], c1 = labc[s * 3 + 1], c2 = labc[s * 3 + 2];
    int l = 0; float best = c0;
    if (c1 > best) { l = 1; best = c1; }
    if (c2 > best) { l = 2; best = c2; }
    if (l == 0) l = (c2 > c1) ? 2 : ((c1 > c2) ? 1 : 0);
    lab[s] = l;
    labeled[s] = (l != 0) ? 1 : 0;
}

// ---------------- row normalize (for affinity) ----------------
__global__ void wesup_norm(const float* __restrict__ mean, float* __restrict__ nrm) {
    const int row = blockIdx.x;
    const int tid = threadIdx.x;
    __shared__ float red[128];
    float ss = 0.0f;
    for (int c = tid; c < C_FEAT; c += 128) {
        float v = mean[(size_t)row * C_FEAT + c];
        ss += v * v;
    }
    red[tid] = ss;
    __syncthreads();
    for (int off = 64; off > 0; off >>= 1) {
        if (tid < off) red[tid] += red[tid + off];
        __syncthreads();
    }
    float inv = 1.0f / fmaxf(sqrtf(red[0]), 1e-12f);
    for (int c = tid; c < C_FEAT; c += 128)
        nrm[(size_t)row * C_FEAT + c] = mean[(size_t)row * C_FEAT + c] * inv;
}

// ---------------- WMMA f32 GEMM: C = relu?(A @ B + bias) ----------------
// One wave computes a 16 x (16*NTW) strip: the A fragment is loaded once per
// K-step and reused across NTW consecutive WMMAs (4x less A traffic, 4
// independent XDL ops per load for latency hiding).
// A: (Mt*16) x K row-major, zero-padded rows.
// TRANS_B=false: B is K x N row-major (ldb = N). TRANS_B=true: B is (Nt*16) x K
// row-major and used as B^T (affinity: nrm @ nrm^T).
template<bool TRANS_B, bool RELU, bool HAS_BIAS, int NTW>
__global__ void wesup_gemm16(const float* __restrict__ A, const float* __restrict__ B,
                             const float* __restrict__ bias, float* __restrict__ C,
                             int Mt, int Nt, int K, int ldb) {
    const int wave = threadIdx.x >> 5;
    const int lane = threadIdx.x & 31;
    const int ntg  = Nt / NTW;                       // N-tile groups
    const int tile = blockIdx.x * (blockDim.x >> 5) + wave;
    if (tile >= Mt * ntg) return;
    const int mt  = tile / ntg;
    const int nt0 = (tile % ntg) * NTW;
    const int hi  = lane >> 4;   // 0: lanes 0-15, 1: lanes 16-31
    const int idx = lane & 15;
    const int arow = mt * 16 + idx;                  // A row held by this lane

    const float* __restrict__ aP = A + (size_t)arow * K;
    const float* bT[NTW];
    if (TRANS_B) {
        #pragma unroll
        for (int i = 0; i < NTW; ++i)
            bT[i] = B + (size_t)((nt0 + i) * 16 + idx) * K;
    }

    v8f acc[NTW];
    #pragma unroll
    for (int i = 0; i < NTW; ++i) acc[i] = (v8f){};

    for (int k0 = 0; k0 < K; k0 += 4) {
        const int ka = k0 + hi * 2;  // v0 holds K rows {k0,k0+2}, v1 holds {k0+1,k0+3}
        v2f a;
        a.x = aP[ka];
        a.y = aP[ka + 1];
        #pragma unroll
        for (int i = 0; i < NTW; ++i) {
            v2f b;
            if (TRANS_B) {
                b.x = bT[i][ka];
                b.y = bT[i][ka + 1];
            } else {
                const int col = (nt0 + i) * 16 + idx;
                b.x = B[(size_t)ka * ldb + col];
                b.y = B[(size_t)(ka + 1) * ldb + col];
            }
            acc[i] = __builtin_amdgcn_wmma_f32_16x16x4_f32(false, a, false, b,
                                                           (short)0, acc[i], false, false);
        }
    }

    const int N = Nt * 16;
    #pragma unroll
    for (int i = 0; i < NTW; ++i) {
        const int col = (nt0 + i) * 16 + idx;
        const float bv = HAS_BIAS ? bias[col] : 0.0f;
        #pragma unroll
        for (int r = 0; r < 8; ++r) {
            int orow = mt * 16 + r + hi * 8;         // C/D: VGPR r holds rows r and r+8
            float v = acc[i][r] + bv;
            if (RELU) v = fmaxf(v, 0.0f);
            C[(size_t)orow * N + col] = v;
        }
    }
}

// ---------------- classifier head + softmax, writes preds into d_out ----------------
__global__ void wesup_head(const float* __restrict__ h3, const float* __restrict__ wc,
                           const float* __restrict__ bc, float* __restrict__ out) {
    int s = blockIdx.x * blockDim.x + threadIdx.x;
    if (s >= NUM_SEG) return;
    float l0 = bc[0], l1 = bc[1];
    #pragma unroll
    for (int k = 0; k < 32; ++k) {
        float v = h3[s * 32 + k];
        l0 += v * wc[k * 2 + 0];
        l1 += v * wc[k * 2 + 1];
    }
    float m = fmaxf(l0, l1);
    float e0 = __expf(l0 - m), e1 = __expf(l1 - m);
    float inv = 1.0f / (e0 + e1);
    out[s * 2 + 0] = e0 * inv;
    out[s * 2 + 1] = e1 * inv;
}

// ---------------- masked row max/argmax over affinity ----------------
__global__ void wesup_best(const float* __restrict__ aff, const int* __restrict__ labeled,
                           float* __restrict__ best, int* __restrict__ bidx) {
    int s = blockIdx.x * blockDim.x + threadIdx.x;
    if (s >= NUM_SEG) return;
    float bv = -INFINITY; int bi = 0;
    const float* row = aff + (size_t)s * M_PAD;
    for (int j = 0; j < NUM_SEG; ++j) {
        float v = labeled[j] ? row[j] : -INFINITY;
        if (v > bv) { bv = v; bi = j; }   // first-max like jnp.argmax
    }
    best[s] = bv;
    bidx[s] = bi;
}

// ---------------- propagate labels + write remaining outputs ----------------
__global__ void wesup_final(const int* __restrict__ lab, const int* __restrict__ labeled,
                            const float* __restrict__ best, const int* __restrict__ bidx,
                            float* __restrict__ out) {
    int s = blockIdx.x * blockDim.x + threadIdx.x;
    if (s >= NUM_SEG) return;
    int l = lab[s];
    int ld = labeled[s];
    int nl = (!ld && best[s] >= 0.97f) ? lab[bidx[s]] : l;
    out[NUM_SEG * 2 + s]               = (float)l;
    out[NUM_SEG * 2 + NUM_SEG + s]     = (float)ld;
    out[NUM_SEG * 2 + 2 * NUM_SEG + s] = (float)nl;
}

extern "C" void kernel_launch(void* const* d_in, const int* in_sizes, int n_in,
                              void* d_out, int out_size, void* d_ws, size_t ws_size,
                              hipStream_t stream) {
    const float* fm = (const float*)d_in[0];   // (2112, 400, 400)
    const int*   sp = (const int*)d_in[1];     // (400, 400)
    const int*   y  = (const int*)d_in[2];     // (400, 400)
    const float* w1 = (const float*)d_in[3];   // (2112, 1024)
    const float* b1 = (const float*)d_in[4];
    const float* w2 = (const float*)d_in[5];   // (1024, 1024)
    const float* b2 = (const float*)d_in[6];
    const float* w3 = (const float*)d_in[7];   // (1024, 32)
    const float* b3 = (const float*)d_in[8];
    const float* wc = (const float*)d_in[9];   // (32, 2)
    const float* bc = (const float*)d_in[10];
    float* out = (float*)d_out;

    float* ws     = (float*)d_ws;
    float* sums   = ws + OFF_SUMS;
    float* cnt    = ws + OFF_CNT;
    float* labc   = ws + OFF_LABC;
    float* mean   = ws + OFF_MEAN;
    float* nrm    = ws + OFF_NRM;
    float* h1     = ws + OFF_H1;
    float* h2     = ws + OFF_H2;
    float* h3     = ws + OFF_H3;
    float* aff    = ws + OFF_AFF;
    float* best   = ws + OFF_BEST;
    int*   bidx   = (int*)(ws + OFF_BIDX);
    int*   lab    = (int*)(ws + OFF_LAB);
    int*   labd   = (int*)(ws + OFF_LBD);

    // 1) zero accumulators (sums + cnt + label counts)
    {
        int n = OFF_MEAN;   // everything before mean buffer
        wesup_zero<<<(n + 255) / 256, 256, 0, stream>>>(ws, n);
    }
    // 2) segment sums of features (memory-bound main pass: 1.35 GB streamed)
    {
        dim3 grid(C_FEAT / CH_PER_BLK, PIX_CHUNKS);
        wesup_segsum<<<grid, 256, 0, stream>>>(fm, sp, sums);
    }
    // 3) pixel counts + label votes
    wesup_counts<<<64, 256, 0, stream>>>(sp, y, cnt, labc);
    // 4) mean features (padded 544 x 2112)
    {
        int n = M_PAD * C_FEAT;
        wesup_mean<<<(n + 255) / 256, 256, 0, stream>>>(sums, cnt, mean);
    }
    // 5) majority labels
    wesup_labels<<<(NUM_SEG + 255) / 256, 256, 0, stream>>>(labc, lab, labd);
    // 6) normalized rows for affinity
    wesup_norm<<<M_PAD, 128, 0, stream>>>(mean, nrm);

    // 7) MLP via f32 WMMA (each wave: 16x64 strip, 8 waves/block)
    {
        int tiles = M_TILES * (64 / 4);   // N=1024, NTW=4
        wesup_gemm16<false, true, true, 4><<<(tiles + 7) / 8, 256, 0, stream>>>(
            mean, w1, b1, h1, M_TILES, 64, C_FEAT, 1024);
    }
    {
        int tiles = M_TILES * (64 / 4);
        wesup_gemm16<false, true, true, 4><<<(tiles + 7) / 8, 256, 0, stream>>>(
            h1, w2, b2, h2, M_TILES, 64, 1024, 1024);
    }
    {
        int tiles = M_TILES * (2 / 2);    // N=32, NTW=2
        wesup_gemm16<false, true, true, 2><<<(tiles + 7) / 8, 256, 0, stream>>>(
            h2, w3, b3, h3, M_TILES, 2, 1024, 32);
    }
    // 8) classifier head + softmax -> preds in d_out
    wesup_head<<<(NUM_SEG + 255) / 256, 256, 0, stream>>>(h3, wc, bc, out);

    // 9) affinity aff = nrm @ nrm^T via WMMA (544x544 padded, NTW=2: 34/2=17)
    {
        int tiles = M_TILES * (M_TILES / 2);
        wesup_gemm16<true, false, false, 2><<<(tiles + 7) / 8, 256, 0, stream>>>(
            nrm, nrm, nullptr, aff, M_TILES, M_TILES, C_FEAT, 0);
    }
    // 10) masked max/argmax per row
    wesup_best<<<(NUM_SEG + 255) / 256, 256, 0, stream>>>(aff, labd, best, bidx);
    // 11) propagate + write lab/labeled/new_lab
    wesup_final<<<(NUM_SEG + 255) / 256, 256, 0, stream>>>(lab, labd, best, bidx, out);
}